// snn_input_41583873360169
// MI455X (gfx1250) — compile-verified
//
#include <hip/hip_runtime.h>

// CDNA5 / gfx1250, wave32. Pure bandwidth-bound elementwise SNN update:
//   m' = mempot + image; spike = (m' >= 1); m' -= spike; traces = 0
// 640 MB traffic @ 23.3 TB/s => ~27.5us floor. Strategy: b128 vector memory
// ops with non-temporal (TH_NT) cache policy on the streaming data (working
// set >> 192MB L2, zero reuse).

typedef __attribute__((ext_vector_type(4))) float v4f;

#define MEMPOT_THRES 1.0f

__global__ __launch_bounds__(256) void snn_vec4_kernel(
    const v4f* __restrict__ image,
    const v4f* __restrict__ mempot,
    float* __restrict__ out,      // [spikes | traces | mempot_new], each n floats
    long long n)                  // total element count (n % 4 == 0 portion handled here)
{
    long long i = (long long)blockIdx.x * blockDim.x + threadIdx.x;
    long long n4 = n >> 2;
    if (i >= n4) return;

    v4f img = __builtin_nontemporal_load(image + i);
    v4f mp  = __builtin_nontemporal_load(mempot + i);

    v4f m = mp + img;

    v4f spk, mo;
#pragma unroll
    for (int j = 0; j < 4; ++j) {
        float mj  = m[j];
        float sel = (mj >= MEMPOT_THRES) ? 1.0f : 0.0f;  // v_cmp + v_cndmask
        spk[j] = sel;
        mo[j]  = mj - sel * MEMPOT_THRES;                // branchless reset
    }

    v4f zero = {0.0f, 0.0f, 0.0f, 0.0f};

    v4f* spikes_v = (v4f*)(out);                 // [0, n)
    v4f* traces_v = (v4f*)(out + n);             // [n, 2n)
    v4f* mpout_v  = (v4f*)(out + 2 * n);         // [2n, 3n)

    __builtin_nontemporal_store(spk,  spikes_v + i);
    __builtin_nontemporal_store(zero, traces_v + i);
    __builtin_nontemporal_store(mo,   mpout_v  + i);
}

// Scalar tail for n % 4 != 0 (defensive; N_NEU = 2^25 is divisible by 4).
__global__ void snn_tail_kernel(
    const float* __restrict__ image,
    const float* __restrict__ mempot,
    float* __restrict__ out,
    long long n, long long start)
{
    long long i = start + (long long)blockIdx.x * blockDim.x + threadIdx.x;
    if (i >= n) return;
    float m   = mempot[i] + image[i];
    float sel = (m >= MEMPOT_THRES) ? 1.0f : 0.0f;
    out[i]         = sel;                      // spikes
    out[n + i]     = 0.0f;                     // traces
    out[2 * n + i] = m - sel * MEMPOT_THRES;   // mempot_new
}

extern "C" void kernel_launch(void* const* d_in, const int* in_sizes, int n_in,
                              void* d_out, int out_size, void* d_ws, size_t ws_size,
                              hipStream_t stream) {
    (void)n_in; (void)out_size; (void)d_ws; (void)ws_size;

    const float* image  = (const float*)d_in[0];
    const float* mempot = (const float*)d_in[1];
    float* out = (float*)d_out;

    long long n  = (long long)in_sizes[0];
    long long n4 = n >> 2;               // float4 chunks
    long long rem_start = n4 << 2;

    if (n4 > 0) {
        const int block = 256;           // 8 wave32s
        long long grid = (n4 + block - 1) / block;
        snn_vec4_kernel<<<(dim3)(unsigned)grid, block, 0, stream>>>(
            (const v4f*)image, (const v4f*)mempot, out, n);
    }
    if (rem_start < n) {
        long long rem = n - rem_start;
        const int block = 64;
        long long grid = (rem + block - 1) / block;
        snn_tail_kernel<<<(dim3)(unsigned)grid, block, 0, stream>>>(
            image, mempot, out, n, rem_start);
    }
}